// GATBlock_7919919693916
// MI455X (gfx1250) — compile-verified
//
#include <hip/hip_runtime.h>
#include <math.h>

typedef float v2f __attribute__((ext_vector_type(2)));
typedef float v8f __attribute__((ext_vector_type(8)));

#define NEG_SLOPE 0.2f

__device__ __forceinline__ void atomic_max_float(float* addr, float val) {
  // classic monotonic-bit-pattern trick; works for mixed signs with -inf init
  if (val >= 0.0f) atomicMax((int*)addr, __float_as_int(val));
  else             atomicMin((unsigned int*)addr, __float_as_uint(val));
}

// ---------------------------------------------------------------------------
// GEMM: C[M,N] = A[M,K] @ B[K,N] (+ optional bias[N]) using V_WMMA_F32_16X16X4_F32
// One wave32 per 16x16 output tile. M,N multiples of 16; K multiple of 4.
// A-frag (16x4): lane L holds rows m=L&15, k = (L>>4)*2 + {0,1}
// B-frag (4x16): lane L holds col n=L&15, k = (L>>4)*2 + {0,1}
// D (16x16):     lane L, elem v -> row (L>>4)*8 + v, col n=L&15
// ---------------------------------------------------------------------------
__global__ __launch_bounds__(256) void gemm_wmma_f32(
    const float* __restrict__ A, const float* __restrict__ B,
    const float* __restrict__ bias, float* __restrict__ C,
    int M, int K, int N) {
  const int lane = threadIdx.x & 31;
  const int wave = blockIdx.x * (blockDim.x >> 5) + (threadIdx.x >> 5);
  const int tilesN = N >> 4;
  const int tilesM = M >> 4;
  if (wave >= tilesM * tilesN) return;   // wave-uniform: EXEC stays all-1 below
  const int tm = wave / tilesN;
  const int tn = wave - tm * tilesN;
  const int mrow  = tm * 16 + (lane & 15);
  const int ncol  = tn * 16 + (lane & 15);
  const int khalf = (lane >> 4) << 1;    // 0 or 2

  v8f acc = {};
  for (int k = 0; k < K; k += 4) {
    const float* ap = A + (size_t)mrow * N * 0 + (size_t)mrow * K + (k + khalf);
    v2f a; a.x = ap[0]; a.y = ap[1];
    const float* bp = B + (size_t)(k + khalf) * N + ncol;
    v2f b; b.x = bp[0]; b.y = bp[N];
    acc = __builtin_amdgcn_wmma_f32_16x16x4_f32(
        /*neg_a=*/false, a, /*neg_b=*/false, b,
        /*c_mod=*/(short)0, acc, /*reuse_a=*/false, /*reuse_b=*/false);
  }

  const int mbase = tm * 16 + ((lane >> 4) << 3);
  const float bv = bias ? bias[ncol] : 0.0f;
#pragma unroll
  for (int v = 0; v < 8; ++v)
    C[(size_t)(mbase + v) * N + ncol] = acc[v] + bv;
}

// ---------------------------------------------------------------------------
// Small init kernels
// ---------------------------------------------------------------------------
__global__ void fill_value(float* __restrict__ p, float v, long n) {
  long t = (long)blockIdx.x * blockDim.x + threadIdx.x;
  if (t < n) p[t] = v;
}

__global__ void fill_rows_bias(float* __restrict__ out, const float* __restrict__ bias,
                               long rows, int C) {
  long t = (long)blockIdx.x * blockDim.x + threadIdx.x;
  if (t < rows * (long)C) out[t] = bias[(int)(t % C)];
}

// ---------------------------------------------------------------------------
// Attention logits per node: as[i,h] = <h[i,h,:], att_src[h,:]>, same for ad
// ---------------------------------------------------------------------------
__global__ void att_scores(const float* __restrict__ h,
                           const float* __restrict__ att_src,
                           const float* __restrict__ att_dst,
                           float* __restrict__ as, float* __restrict__ ad,
                           int n, int H, int C) {
  long t = (long)blockIdx.x * blockDim.x + threadIdx.x;  // (node*H + head)
  if (t >= (long)n * H) return;
  const int head = (int)(t % H);
  const float4* hv = (const float4*)(h + (size_t)t * C);
  const float4* sv = (const float4*)(att_src + (size_t)head * C);
  const float4* dv = (const float4*)(att_dst + (size_t)head * C);
  float s = 0.f, d = 0.f;
  for (int i = 0; i < (C >> 2); ++i) {
    float4 xv = hv[i], a = sv[i], b = dv[i];
    s += xv.x * a.x + xv.y * a.y + xv.z * a.z + xv.w * a.w;
    d += xv.x * b.x + xv.y * b.y + xv.z * b.z + xv.w * b.w;
  }
  as[t] = s; ad[t] = d;
}

// ---------------------------------------------------------------------------
// Edge pass A: e = leaky_relu(a_src[src]+a_dst[dst]); segment max into m[dst]
// Edges e<E come from edge_index, e>=E are self loops.
// ---------------------------------------------------------------------------
__global__ void edge_logits_max(const int* __restrict__ ei,
                                const float* __restrict__ as,
                                const float* __restrict__ ad,
                                float* __restrict__ e, float* __restrict__ m,
                                int E, int n, int H) {
  long t = (long)blockIdx.x * blockDim.x + threadIdx.x;
  long total = (long)(E + n) * H;
  if (t >= total) return;
  const int edge = (int)(t / H);
  const int hd   = (int)(t % H);
  const int src = (edge < E) ? ei[edge]     : edge - E;
  const int dst = (edge < E) ? ei[E + edge] : edge - E;
  float v = as[(size_t)src * H + hd] + ad[(size_t)dst * H + hd];
  v = (v > 0.f) ? v : NEG_SLOPE * v;
  e[t] = v;
  atomic_max_float(&m[(size_t)dst * H + hd], v);
}

// Edge pass B: e <- exp(e - m[dst]); segment sum into s[dst]
__global__ void edge_exp_sum(const int* __restrict__ ei, float* __restrict__ e,
                             const float* __restrict__ m, float* __restrict__ s,
                             int E, int n, int H) {
  long t = (long)blockIdx.x * blockDim.x + threadIdx.x;
  long total = (long)(E + n) * H;
  if (t >= total) return;
  const int edge = (int)(t / H);
  const int hd   = (int)(t % H);
  const int dst = (edge < E) ? ei[E + edge] : edge - E;
  float v = expf(e[t] - m[(size_t)dst * H + hd]);
  e[t] = v;
  atomicAdd(&s[(size_t)dst * H + hd], v);
}

// Edge pass C: out[dst,h,c] += (e[edge,h] / (s[dst,h]+eps)) * feat[src,h,c]
// Flat over (E')*H*C so 256 consecutive lanes hit one contiguous row (coalesced).
__global__ void edge_scatter(const int* __restrict__ ei, const float* __restrict__ e,
                             const float* __restrict__ s, const float* __restrict__ feat,
                             float* __restrict__ out, int E, int n, int H, int C) {
  long t = (long)blockIdx.x * blockDim.x + threadIdx.x;
  long total = (long)(E + n) * H * C;
  if (t >= total) return;
  const int  c    = (int)(t % C);
  const long eh   = t / C;
  const int  hd   = (int)(eh % H);
  const int  edge = (int)(eh / H);
  const int src = (edge < E) ? ei[edge]     : edge - E;
  const int dst = (edge < E) ? ei[E + edge] : edge - E;
  const float alpha = e[eh] / (s[(size_t)dst * H + hd] + 1e-16f);
  atomicAdd(&out[((size_t)dst * H + hd) * C + c],
            alpha * feat[((size_t)src * H + hd) * C + c]);
}

// ---------------------------------------------------------------------------
// LayerNorm (+ optional residual add) + ELU, one wave32 per row
// ---------------------------------------------------------------------------
__global__ void ln_add_elu(const float* __restrict__ x, const float* __restrict__ g,
                           const float* __restrict__ b, const float* __restrict__ add,
                           float* __restrict__ y, int n, int C) {
  const int row  = blockIdx.x * (blockDim.x >> 5) + (threadIdx.x >> 5);
  const int lane = threadIdx.x & 31;
  if (row >= n) return;
  const float* xr = x + (size_t)row * C;
  float s = 0.f, ss = 0.f;
  for (int c = lane; c < C; c += 32) { float v = xr[c]; s += v; ss += v * v; }
  for (int off = 16; off; off >>= 1) {
    s  += __shfl_xor(s,  off, 32);
    ss += __shfl_xor(ss, off, 32);
  }
  const float mu  = s / (float)C;
  const float var = ss / (float)C - mu * mu;
  const float inv = rsqrtf(var + 1e-5f);
  const float* ar = add ? add + (size_t)row * C : nullptr;
  float* yr = y + (size_t)row * C;
  for (int c = lane; c < C; c += 32) {
    float v = (xr[c] - mu) * inv * g[c] + b[c];
    if (ar) v += ar[c];
    yr[c] = (v > 0.f) ? v : expm1f(v);
  }
}

// ---------------------------------------------------------------------------
extern "C" void kernel_launch(void* const* d_in, const int* in_sizes, int n_in,
                              void* d_out, int out_size, void* d_ws, size_t ws_size,
                              hipStream_t stream) {
  const float* x    = (const float*)d_in[0];
  const int*   ei   = (const int*)  d_in[1];
  const float* W1   = (const float*)d_in[2];
  const float* aS1  = (const float*)d_in[3];
  const float* aD1  = (const float*)d_in[4];
  const float* b1   = (const float*)d_in[5];
  const float* W2   = (const float*)d_in[6];
  const float* aS2  = (const float*)d_in[7];
  const float* aD2  = (const float*)d_in[8];
  const float* b2   = (const float*)d_in[9];
  const float* g1   = (const float*)d_in[10];
  const float* be1  = (const float*)d_in[11];
  const float* g2   = (const float*)d_in[12];
  const float* be2  = (const float*)d_in[13];
  const float* pW   = (const float*)d_in[14];
  const float* pb   = (const float*)d_in[15];

  const int IN_CH = 128, HEADS = 4, OUT_CH = 64, HC = HEADS * OUT_CH; // 256
  const int N  = in_sizes[0] / IN_CH;
  const int E  = in_sizes[1] / 2;
  const long EP = (long)E + N;   // edges + self loops

  // workspace layout (f32)
  float* ws   = (float*)d_ws;
  float* h1   = ws; ws += (size_t)N * HC;
  float* out1 = ws; ws += (size_t)N * HC;      // becomes h2 after in-place LN+ELU
  float* idb  = ws; ws += (size_t)N * OUT_CH;  // identity = x@pW + pb
  float* z2   = ws; ws += (size_t)N * OUT_CH;  // layer-2 features
  float* out2 = ws; ws += (size_t)N * OUT_CH;
  float* as1  = ws; ws += (size_t)N * HEADS;
  float* ad1  = ws; ws += (size_t)N * HEADS;
  float* m1   = ws; ws += (size_t)N * HEADS;
  float* s1   = ws; ws += (size_t)N * HEADS;
  float* e1   = ws; ws += (size_t)EP * HEADS;
  float* as2  = ws; ws += (size_t)N;
  float* ad2  = ws; ws += (size_t)N;
  float* m2   = ws; ws += (size_t)N;
  float* s2   = ws; ws += (size_t)N;
  float* e2   = ws; ws += (size_t)EP;

  auto cdiv = [](long a, long b) -> int { return (int)((a + b - 1) / b); };
  const int BLK = 256;
  const int WPB = BLK / 32;   // waves per block

  // ---- GEMMs (WMMA f32 16x16x4): h1 = x@W1 ; identity = x@pW + pb
  {
    long waves = ((long)N / 16) * (HC / 16);
    gemm_wmma_f32<<<cdiv(waves, WPB), BLK, 0, stream>>>(x, W1, nullptr, h1, N, IN_CH, HC);
  }
  {
    long waves = ((long)N / 16) * (OUT_CH / 16);
    gemm_wmma_f32<<<cdiv(waves, WPB), BLK, 0, stream>>>(x, pW, pb, idb, N, IN_CH, OUT_CH);
  }

  // ---- Layer 1 attention
  att_scores<<<cdiv((long)N * HEADS, BLK), BLK, 0, stream>>>(h1, aS1, aD1, as1, ad1, N, HEADS, OUT_CH);
  fill_value<<<cdiv((long)N * HEADS, BLK), BLK, 0, stream>>>(m1, -INFINITY, (long)N * HEADS);
  fill_value<<<cdiv((long)N * HEADS, BLK), BLK, 0, stream>>>(s1, 0.0f,      (long)N * HEADS);
  fill_rows_bias<<<cdiv((long)N * HC, BLK), BLK, 0, stream>>>(out1, b1, N, HC);

  edge_logits_max<<<cdiv(EP * HEADS, BLK), BLK, 0, stream>>>(ei, as1, ad1, e1, m1, E, N, HEADS);
  edge_exp_sum  <<<cdiv(EP * HEADS, BLK), BLK, 0, stream>>>(ei, e1, m1, s1, E, N, HEADS);
  edge_scatter  <<<cdiv(EP * HEADS * OUT_CH, BLK), BLK, 0, stream>>>(ei, e1, s1, h1, out1, E, N, HEADS, OUT_CH);

  // out1 <- elu(layernorm(out1))  (in-place; becomes h2)
  ln_add_elu<<<cdiv(N, WPB), BLK, 0, stream>>>(out1, g1, be1, nullptr, out1, N, HC);

  // ---- Layer 2: z2 = h2 @ W2
  {
    long waves = ((long)N / 16) * (OUT_CH / 16);
    gemm_wmma_f32<<<cdiv(waves, WPB), BLK, 0, stream>>>(out1, W2, nullptr, z2, N, HC, OUT_CH);
  }

  att_scores<<<cdiv((long)N, BLK), BLK, 0, stream>>>(z2, aS2, aD2, as2, ad2, N, 1, OUT_CH);
  fill_value<<<cdiv((long)N, BLK), BLK, 0, stream>>>(m2, -INFINITY, (long)N);
  fill_value<<<cdiv((long)N, BLK), BLK, 0, stream>>>(s2, 0.0f,      (long)N);
  fill_rows_bias<<<cdiv((long)N * OUT_CH, BLK), BLK, 0, stream>>>(out2, b2, N, OUT_CH);

  edge_logits_max<<<cdiv(EP, BLK), BLK, 0, stream>>>(ei, as2, ad2, e2, m2, E, N, 1);
  edge_exp_sum  <<<cdiv(EP, BLK), BLK, 0, stream>>>(ei, e2, m2, s2, E, N, 1);
  edge_scatter  <<<cdiv(EP * OUT_CH, BLK), BLK, 0, stream>>>(ei, e2, s2, z2, out2, E, N, 1, OUT_CH);

  // ---- final: d_out = elu(layernorm(out2) + identity)
  ln_add_elu<<<cdiv(N, WPB), BLK, 0, stream>>>(out2, g2, be2, idb, (float*)d_out, N, OUT_CH);

  (void)n_in; (void)out_size; (void)ws_size;
}